// PointNetFeaturePropagation1_15238543966701
// MI455X (gfx1250) — compile-verified
//
#include <hip/hip_runtime.h>
#include <hip/hip_bf16.h>

typedef float v2f __attribute__((ext_vector_type(2)));
typedef float v8f __attribute__((ext_vector_type(8)));

#define BATCH 8
#define NPTS  8192
#define SPTS  2048
#define DFEAT 256
#define CIN   512
#define C0    256
#define C1    256
#define NN_EPS 1e-8f
#define BN_EPS 1e-5f

#if __has_builtin(__builtin_amdgcn_global_load_async_to_lds_b32)
#define HAVE_ASYNC_LDS 1
typedef __attribute__((address_space(1))) int GlobalInt;
typedef __attribute__((address_space(3))) int LdsInt;
#else
#define HAVE_ASYNC_LDS 0
#endif

// ---------------------------------------------------------------------------
// Kernel 1: transpose points2 [B,D,S] -> p2t [B,S,D] (tiled via LDS).
// Uses the CDNA5 async global->LDS path (ASYNCcnt) when the builtin exists.
// ---------------------------------------------------------------------------
__global__ __launch_bounds__(256) void transpose_kernel(
    const float* __restrict__ in, float* __restrict__ out, int D, int S) {
  __shared__ float tile[32][33];
  int b  = blockIdx.z;
  int d0 = blockIdx.y * 32;
  int s0 = blockIdx.x * 32;
  int tx = threadIdx.x & 31;
  int ty = threadIdx.x >> 5;  // 8 rows
  const float* src = in + (size_t)b * D * S;
#if HAVE_ASYNC_LDS
  for (int i = 0; i < 32; i += 8)
    __builtin_amdgcn_global_load_async_to_lds_b32(
        (GlobalInt*)(src + (size_t)(d0 + ty + i) * S + s0 + tx),
        (LdsInt*)&tile[ty + i][tx], 0, 0);
#if __has_builtin(__builtin_amdgcn_s_wait_asynccnt)
  __builtin_amdgcn_s_wait_asynccnt(0);
#else
  asm volatile("s_wait_asynccnt 0x0" ::: "memory");
#endif
#else
  for (int i = 0; i < 32; i += 8)
    tile[ty + i][tx] = src[(size_t)(d0 + ty + i) * S + s0 + tx];
#endif
  __syncthreads();
  float* dst = out + (size_t)b * S * D;
  for (int i = 0; i < 32; i += 8)
    dst[(size_t)(s0 + ty + i) * D + d0 + tx] = tile[tx][ty + i];
}

// ---------------------------------------------------------------------------
// Kernel 2: 3-NN search. xyz1 [B,3,N], xyz2 [B,3,S] (staged in LDS).
// One thread per query point; running top-3 insertion (stable-tie order).
// ---------------------------------------------------------------------------
__global__ __launch_bounds__(256) void three_nn_kernel(
    const float* __restrict__ xyz1, const float* __restrict__ xyz2,
    int* __restrict__ nn_idx, float* __restrict__ nn_w, int N, int S) {
  __shared__ float sx[SPTS], sy[SPTS], sz[SPTS];
  int b = blockIdx.y;
  int n = blockIdx.x * blockDim.x + threadIdx.x;
  const float* x2 = xyz2 + (size_t)b * 3 * S;
  for (int s = threadIdx.x; s < S; s += blockDim.x) {
    sx[s] = x2[s];
    sy[s] = x2[S + s];
    sz[s] = x2[2 * S + s];
  }
  __syncthreads();
  const float* x1 = xyz1 + (size_t)b * 3 * N;
  float qx = x1[n], qy = x1[N + n], qz = x1[2 * N + n];
  float d0 = 3.0e38f, d1 = 3.0e38f, d2 = 3.0e38f;
  int i0 = 0, i1 = 0, i2 = 0;
  for (int s = 0; s < S; ++s) {
    float dx = qx - sx[s], dy = qy - sy[s], dz = qz - sz[s];
    float d = fmaf(dx, dx, fmaf(dy, dy, dz * dz));
    if (d < d2) {
      if (d < d1) {
        if (d < d0) { d2 = d1; i2 = i1; d1 = d0; i1 = i0; d0 = d; i0 = s; }
        else        { d2 = d1; i2 = i1; d1 = d;  i1 = s; }
      } else        { d2 = d;  i2 = s; }
    }
  }
  float r0 = 1.0f / (d0 + NN_EPS);
  float r1 = 1.0f / (d1 + NN_EPS);
  float r2 = 1.0f / (d2 + NN_EPS);
  float inv = 1.0f / (r0 + r1 + r2);
  size_t o = ((size_t)b * N + n) * 3;
  nn_idx[o + 0] = i0; nn_idx[o + 1] = i1; nn_idx[o + 2] = i2;
  nn_w[o + 0] = r0 * inv; nn_w[o + 1] = r1 * inv; nn_w[o + 2] = r2 * inv;
}

// ---------------------------------------------------------------------------
// Kernel 3: weighted gather. p2t [B,S,D] + idx/w -> interp [B,D,N].
// Reads coalesced along D; LDS tile transposes so writes along N coalesce.
// ---------------------------------------------------------------------------
__global__ __launch_bounds__(256) void interp_kernel(
    const float* __restrict__ p2t, const int* __restrict__ nn_idx,
    const float* __restrict__ nn_w, float* __restrict__ interp,
    int N, int S, int D) {
  __shared__ float tile[32][33];  // [n_local][d_local]
  __shared__ int   sidx[32][3];
  __shared__ float sw[32][3];
  int b = blockIdx.y, n0 = blockIdx.x * 32;
  int t = threadIdx.x;
  if (t < 96) {
    int nl = t / 3, k = t % 3;
    size_t o = ((size_t)b * N + n0 + nl) * 3 + k;
    sidx[nl][k] = nn_idx[o];
    sw[nl][k]   = nn_w[o];
  }
  __syncthreads();
  int tx = t & 31;   // channel lane
  int ty = t >> 5;   // 8 query rows
  const float* base = p2t + (size_t)b * S * D;
  for (int dd0 = 0; dd0 < D; dd0 += 32) {
    for (int ns = 0; ns < 32; ns += 8) {
      int nl = ns + ty;
      float v = sw[nl][0] * base[(size_t)sidx[nl][0] * D + dd0 + tx];
      v = fmaf(sw[nl][1], base[(size_t)sidx[nl][1] * D + dd0 + tx], v);
      v = fmaf(sw[nl][2], base[(size_t)sidx[nl][2] * D + dd0 + tx], v);
      tile[nl][tx] = v;
    }
    __syncthreads();
    for (int ds = 0; ds < 32; ds += 8) {
      int dl = ds + ty;
      interp[((size_t)b * D + dd0 + dl) * N + n0 + tx] = tile[tx][dl];
    }
    __syncthreads();
  }
}

// ---------------------------------------------------------------------------
// Kernel 4: fp32 WMMA GEMM with compile-time shapes.
// Y[b] = A[COUT,LDA] x X[b][K,N].  Each wave owns a 32x64 output tile:
// 8 v8f accumulators, A-frag pair reused over 4 B-frags, 8x
// v_wmma_f32_16x16x4_f32 per K-quad.  All dims constexpr so every load is
// base + immediate offset (no per-iteration 64-bit address chains).
// ---------------------------------------------------------------------------
template <int LDA, int K, int N>
__device__ __forceinline__ void gemm_phase(
    const float* __restrict__ A, int kofs, const float* __restrict__ act,
    int m0, int n0, int l, int half, v8f (&c)[2][4]) {
  const float* a0 = A + (m0 + l) * LDA + kofs + 2 * half;
  const float* a1 = a0 + 16 * LDA;
  const float* bp = act + (2 * half) * N + n0 + l;
#pragma unroll 8
  for (int k = 0; k < K; k += 4) {
    if ((k & 31) == 0 && (k + 32) < K)
      __builtin_prefetch(bp + (k + 32) * N, 0, 1);
    v2f a[2];
    a[0] = *(const v2f*)(a0 + k);        // A[m0+l   ][k+2h : k+2h+1]
    a[1] = *(const v2f*)(a1 + k);        // A[m0+16+l][k+2h : k+2h+1]
    v2f bb[4];
#pragma unroll
    for (int j = 0; j < 4; ++j) {
      bb[j].x = bp[k * N + 16 * j];      // B[k+2h  ][n0+16j+l]
      bb[j].y = bp[k * N + N + 16 * j];  // B[k+2h+1][n0+16j+l]
    }
#pragma unroll
    for (int j = 0; j < 4; ++j) {
      c[0][j] = __builtin_amdgcn_wmma_f32_16x16x4_f32(
          false, a[0], false, bb[j], (short)0, c[0][j], false, false);
      c[1][j] = __builtin_amdgcn_wmma_f32_16x16x4_f32(
          false, a[1], false, bb[j], (short)0, c[1][j], false, false);
    }
  }
}

template <int LDA, int KK1, int KK2>
__global__ __launch_bounds__(128) void gemm_wmma_kernel(
    const float* __restrict__ A, const float* __restrict__ act1,
    const float* __restrict__ act2, float* __restrict__ Y) {
  constexpr int N = NPTS;
  constexpr int COUT = 256;
  constexpr int tilesN = N / 64;
  constexpr int tilesM = COUT / 32;
  int wid  = blockIdx.x * 4 + (threadIdx.x >> 5);
  int lane = threadIdx.x & 31;
  int l = lane & 15, half = lane >> 4;
  int tn  = wid % tilesN;
  int tmp = wid / tilesN;
  int tm  = tmp % tilesM;
  int b   = tmp / tilesM;
  int m0 = tm * 32, n0 = tn * 64;

  v8f c[2][4] = {};
  gemm_phase<LDA, KK1, N>(A, 0, act1 + (size_t)b * KK1 * N, m0, n0, l, half, c);
  if constexpr (KK2 > 0)  // wave-uniform: EXEC stays all-ones around WMMA
    gemm_phase<LDA, KK2, N>(A, KK1, act2 + (size_t)b * KK2 * N, m0, n0, l, half, c);

  // D layout: VGPR r, half h -> M = m0 + 16i + 8h + r ; N = n0 + 16j + (lane&15)
  float* out = Y + ((size_t)b * COUT + m0) * N + n0;
#pragma unroll
  for (int i = 0; i < 2; ++i)
#pragma unroll
    for (int j = 0; j < 4; ++j)
#pragma unroll
      for (int r = 0; r < 8; ++r)
        out[(16 * i + 8 * half + r) * N + 16 * j + l] = c[i][j][r];
}

// ---------------------------------------------------------------------------
// Kernel 5: BN stats (training mode, stats over batch*N per channel).
// One block per channel; emits fused scale/shift (conv bias is absorbed by
// the mean subtraction, so it is skipped entirely).
// ---------------------------------------------------------------------------
__global__ __launch_bounds__(256) void bn_stats_kernel(
    const float* __restrict__ Y, const float* __restrict__ gamma,
    const float* __restrict__ beta, float* __restrict__ scaleShift,
    int B, int C, int N) {
  __shared__ float ssum[256], ssq[256];
  int c = blockIdx.x, t = threadIdx.x;
  float s = 0.0f, q = 0.0f;
  for (int b = 0; b < B; ++b) {
    const float* row = Y + ((size_t)b * C + c) * N;
    for (int n = t; n < N; n += blockDim.x) {
      float v = row[n];
      s += v;
      q = fmaf(v, v, q);
    }
  }
  ssum[t] = s; ssq[t] = q;
  __syncthreads();
  for (int st = 128; st > 0; st >>= 1) {
    if (t < st) { ssum[t] += ssum[t + st]; ssq[t] += ssq[t + st]; }
    __syncthreads();
  }
  if (t == 0) {
    float inv  = 1.0f / ((float)B * (float)N);
    float mean = ssum[0] * inv;
    float var  = ssq[0] * inv - mean * mean;
    float sc   = gamma[c] * rsqrtf(var + BN_EPS);
    scaleShift[2 * c]     = sc;
    scaleShift[2 * c + 1] = fmaf(-mean, sc, beta[c]);
  }
}

// ---------------------------------------------------------------------------
// Kernel 6: apply BN scale/shift + ReLU (elementwise, grid-stride)
// ---------------------------------------------------------------------------
__global__ __launch_bounds__(256) void bn_relu_kernel(
    const float* __restrict__ Y, const float* __restrict__ scaleShift,
    float* __restrict__ out, int C, int N, size_t total) {
  size_t i = (size_t)blockIdx.x * blockDim.x + threadIdx.x;
  size_t step = (size_t)gridDim.x * blockDim.x;
  for (; i < total; i += step) {
    int c = (int)((i / (size_t)N) % (size_t)C);
    float v = fmaf(Y[i], scaleShift[2 * c], scaleShift[2 * c + 1]);
    out[i] = v > 0.0f ? v : 0.0f;
  }
}

// ---------------------------------------------------------------------------
extern "C" void kernel_launch(void* const* d_in, const int* in_sizes, int n_in,
                              void* d_out, int out_size, void* d_ws, size_t ws_size,
                              hipStream_t stream) {
  const float* xyz1    = (const float*)d_in[0];   // [8,3,8192]
  const float* xyz2    = (const float*)d_in[1];   // [8,3,2048]
  const float* points1 = (const float*)d_in[2];   // [8,256,8192]
  const float* points2 = (const float*)d_in[3];   // [8,256,2048]
  const float* w0      = (const float*)d_in[4];   // [256,512]
  const float* gamma0  = (const float*)d_in[6];
  const float* beta0   = (const float*)d_in[7];
  const float* w1      = (const float*)d_in[8];   // [256,256]
  const float* gamma1  = (const float*)d_in[10];
  const float* beta1   = (const float*)d_in[11];
  (void)in_sizes; (void)n_in; (void)out_size; (void)ws_size;

  // Workspace layout (floats)
  float* p2t        = (float*)d_ws;                               //  4,194,304
  int*   nn_idx     = (int*)(p2t + (size_t)BATCH * SPTS * DFEAT); //    196,608
  float* nn_w       = (float*)(nn_idx + (size_t)BATCH * NPTS * 3);//    196,608
  float* interp     = nn_w + (size_t)BATCH * NPTS * 3;            // 16,777,216
  float* Yb         = interp + (size_t)BATCH * DFEAT * NPTS;      // 16,777,216
  float* Hb         = Yb + (size_t)BATCH * C0 * NPTS;             // 16,777,216
  float* scaleShift = Hb + (size_t)BATCH * C0 * NPTS;             //        512

  const size_t total = (size_t)BATCH * C0 * NPTS;
  const int gemm_blocks = BATCH * (C0 / 32) * (NPTS / 64) / 4;    // 2048

  // 1) transpose points2 -> [B,S,D]
  transpose_kernel<<<dim3(SPTS / 32, DFEAT / 32, BATCH), 256, 0, stream>>>(
      points2, p2t, DFEAT, SPTS);

  // 2) 3-NN search
  three_nn_kernel<<<dim3(NPTS / 256, BATCH), 256, 0, stream>>>(
      xyz1, xyz2, nn_idx, nn_w, NPTS, SPTS);

  // 3) weighted feature gather -> interp [B,256,N]
  interp_kernel<<<dim3(NPTS / 32, BATCH), 256, 0, stream>>>(
      p2t, nn_idx, nn_w, interp, NPTS, SPTS, DFEAT);

  // 4) layer 1 GEMM: w0 x concat(points1, interp)  (K split 256+256)
  gemm_wmma_kernel<CIN, DFEAT, DFEAT><<<gemm_blocks, 128, 0, stream>>>(
      w0, points1, interp, Yb);
  bn_stats_kernel<<<C0, 256, 0, stream>>>(Yb, gamma0, beta0, scaleShift,
                                          BATCH, C0, NPTS);
  bn_relu_kernel<<<8192, 256, 0, stream>>>(Yb, scaleShift, Hb, C0, NPTS, total);

  // 5) layer 2 GEMM: w1 x H  (reuse Yb for pre-BN output)
  gemm_wmma_kernel<C0, C0, 0><<<gemm_blocks, 128, 0, stream>>>(
      w1, Hb, nullptr, Yb);
  bn_stats_kernel<<<C1, 256, 0, stream>>>(Yb, gamma1, beta1, scaleShift,
                                          BATCH, C1, NPTS);
  bn_relu_kernel<<<8192, 256, 0, stream>>>(Yb, scaleShift, (float*)d_out,
                                           C1, NPTS, total);
}